// Self_Correlation_Per_tt_52089363366009
// MI455X (gfx1250) — compile-verified
//
#include <hip/hip_runtime.h>

typedef __attribute__((ext_vector_type(2))) float v2f;
typedef __attribute__((ext_vector_type(8))) float v8f;

#define IMG_STRIDE 68
#define EPSN 1e-12f

__device__ __forceinline__ v8f wmma_f32_16x16x4(v2f a, v2f b, v8f c) {
  // 8 args: (neg_a, A, neg_b, B, c_mod, C, reuse_a, reuse_b)
  return __builtin_amdgcn_wmma_f32_16x16x4_f32(false, a, false, b, (short)0, c,
                                               false, false);
}

// ---------------------------------------------------------------------------
// Stage 1: per (b,c) 64x64 Gram over P=57*57 window positions via f32 WMMA,
// then normalize (diag gives column norms), then 9 order statistics / column.
// One WG of 128 threads (4 waves) per (b,c); each wave owns a 32x32 quadrant.
// ---------------------------------------------------------------------------
__global__ __launch_bounds__(128) void stage1_kernel(
    const float* __restrict__ x, float* __restrict__ pool) {
  __shared__ float img[64 * IMG_STRIDE];  // 64 rows, padded to 68 cols
  __shared__ float gram[64 * 64];
  __shared__ float invn[64];

  const int tid = threadIdx.x;
  const int bc = blockIdx.x;  // b*64 + c
  const float* src = x + (size_t)bc * 4096;

  // zero pad columns 64..67 (B-side reads can touch cols 64..66)
  for (int idx = tid; idx < 64 * 4; idx += 128) {
    img[(idx >> 2) * IMG_STRIDE + 64 + (idx & 3)] = 0.0f;
  }
  // load 64x64 image into LDS
  for (int idx = tid; idx < 4096; idx += 128) {
    img[(idx >> 6) * IMG_STRIDE + (idx & 63)] = src[idx];
  }
  __syncthreads();

  const int lane = tid & 31;
  const int r = lane & 15;
  const int hi = lane >> 4;    // 0: holds k=0,1   1: holds k=2,3
  const int klo = hi * 2;
  const int w = tid >> 5;      // wave 0..3
  const int ma = (w >> 1) * 32, mb = ma + 16;
  const int na = (w & 1) * 32, nb = na + 16;

  // kernel-position offset for matrix row/col index m (m = ki*8+kj)
  const int offA0 = ((ma + r) >> 3) * IMG_STRIDE + ((ma + r) & 7);
  const int offA1 = ((mb + r) >> 3) * IMG_STRIDE + ((mb + r) & 7);
  const int offB0 = ((na + r) >> 3) * IMG_STRIDE + ((na + r) & 7);
  const int offB1 = ((nb + r) >> 3) * IMG_STRIDE + ((nb + r) & 7);

  v8f c00 = {}, c01 = {}, c10 = {}, c11 = {};

  for (int oi = 0; oi < 57; ++oi) {
    const int base = oi * IMG_STRIDE;
#pragma unroll 2
    for (int cj = 0; cj < 14; ++cj) {  // oj in [0,56), all valid
      const int px = base + cj * 4 + klo;
      v2f a0, a1, b0, b1;
      a0.x = img[px + offA0];  a0.y = img[px + 1 + offA0];
      a1.x = img[px + offA1];  a1.y = img[px + 1 + offA1];
      b0.x = img[px + offB0];  b0.y = img[px + 1 + offB0];
      b1.x = img[px + offB1];  b1.y = img[px + 1 + offB1];
      c00 = wmma_f32_16x16x4(a0, b0, c00);
      c01 = wmma_f32_16x16x4(a0, b1, c01);
      c10 = wmma_f32_16x16x4(a1, b0, c10);
      c11 = wmma_f32_16x16x4(a1, b1, c11);
    }
    {  // last chunk: oj = 56+klo+{0,1}; only oj==56 (lanes 0-15, .x) valid
      const int px = base + 56 + klo;
      const bool vx = (klo == 0);
      v2f a0, a1, b0, b1;
      a0.x = vx ? img[px + offA0] : 0.0f;  a0.y = 0.0f;
      a1.x = vx ? img[px + offA1] : 0.0f;  a1.y = 0.0f;
      b0.x = img[px + offB0];  b0.y = img[px + 1 + offB0];
      b1.x = img[px + offB1];  b1.y = img[px + 1 + offB1];
      c00 = wmma_f32_16x16x4(a0, b0, c00);
      c01 = wmma_f32_16x16x4(a0, b1, c01);
      c10 = wmma_f32_16x16x4(a1, b0, c10);
      c11 = wmma_f32_16x16x4(a1, b1, c11);
    }
  }

  // C/D layout: lanes 0-15 -> M = vgpr, lanes 16-31 -> M = vgpr + 8; N = lane&15
#pragma unroll
  for (int v = 0; v < 8; ++v) {
    const int mo = v + hi * 8;
    gram[(ma + mo) * 64 + (na + r)] = c00[v];
    gram[(ma + mo) * 64 + (nb + r)] = c01[v];
    gram[(mb + mo) * 64 + (na + r)] = c10[v];
    gram[(mb + mo) * 64 + (nb + r)] = c11[v];
  }
  __syncthreads();

  if (tid < 64) {
    invn[tid] = 1.0f / fmaxf(sqrtf(gram[tid * 64 + tid]), EPSN);
  }
  __syncthreads();

  // corr[m,n] = G[m,n] * inv[m] * inv[n] / P, in place
  for (int idx = tid; idx < 4096; idx += 128) {
    const int m = idx >> 6, n = idx & 63;
    gram[idx] = gram[idx] * invn[m] * invn[n] * (1.0f / 3249.0f);
  }
  __syncthreads();

  // order statistics: 2 threads per column, counting rank with tie-break
  int rk[9];
#pragma unroll
  for (int j = 0; j < 9; ++j) rk[j] = (int)rintf(1.0f + 7.75f * (float)j);
  const int n = tid & 63;
  const int i0 = (tid >> 6) * 32;
  for (int i = i0; i < i0 + 32; ++i) {
    const float vi = gram[i * 64 + n];
    int cnt = 0;
    for (int j = 0; j < 64; ++j) {
      const float vj = gram[j * 64 + n];
      cnt += (vj > vi) || (vj == vi && j < i);  // descending rank
    }
#pragma unroll
    for (int jj = 0; jj < 9; ++jj)
      if (cnt == rk[jj]) pool[(size_t)bc * 576 + jj * 64 + n] = vi;
  }
}

// ---------------------------------------------------------------------------
// Stage 2a: per-position channel norms: inv2[b][p] = 1/max(||pool[b][:,p]||,eps)
// ---------------------------------------------------------------------------
__global__ __launch_bounds__(256) void stage2_norm_kernel(
    const float* __restrict__ pool, float* __restrict__ inv2) {
  const int g = blockIdx.x * 256 + threadIdx.x;
  if (g >= 8 * 576) return;
  const int b = g / 576, p = g % 576;
  const float* base = pool + (size_t)b * 64 * 576 + p;
  float s = 0.0f;
#pragma unroll 4
  for (int c = 0; c < 64; ++c) {
    const float v = base[c * 576];
    s += v * v;
  }
  inv2[g] = 1.0f / fmaxf(sqrtf(s), EPSN);
}

// ---------------------------------------------------------------------------
// Stage 2b: per-b 576x576 Gram over K=64 channels via f32 WMMA.
// Grid: 8 b x 9 column strips (64 cols). 4 waves, one 16-col tile each;
// B fragment (16 x v2f) hoisted across the 36 row tiles. Stores transposed.
// ---------------------------------------------------------------------------
__global__ __launch_bounds__(128) void stage2_gram_kernel(
    const float* __restrict__ pool, const float* __restrict__ inv2,
    float* __restrict__ corr2T) {
  const int bid = blockIdx.x;  // 8*9
  const int b = bid / 9, strip = bid % 9;
  const int tid = threadIdx.x;
  const int lane = tid & 31, r = lane & 15, hi = lane >> 4, klo = hi * 2;
  const int w = tid >> 5;
  const int q = (strip * 4 + w) * 16 + r;
  const float* pb = pool + (size_t)b * 64 * 576;
  const float* ib = inv2 + b * 576;

  v2f B[16];
#pragma unroll
  for (int kc = 0; kc < 16; ++kc) {
    const int c0 = kc * 4 + klo;
    B[kc].x = pb[c0 * 576 + q];
    B[kc].y = pb[(c0 + 1) * 576 + q];
  }
  const float invq = ib[q];

  for (int rt = 0; rt < 36; ++rt) {
    const int p0 = rt * 16 + r;
    v8f acc = {};
#pragma unroll
    for (int kc = 0; kc < 16; ++kc) {
      const int c0 = kc * 4 + klo;
      v2f a;
      a.x = pb[c0 * 576 + p0];
      a.y = pb[(c0 + 1) * 576 + p0];
      acc = wmma_f32_16x16x4(a, B[kc], acc);
    }
#pragma unroll
    for (int v = 0; v < 8; ++v) {
      const int p = rt * 16 + v + hi * 8;
      corr2T[((size_t)b * 576 + q) * 576 + p] =
          acc[v] * ib[p] * invq * (1.0f / 64.0f);
    }
  }
}

// ---------------------------------------------------------------------------
// Stage 2c: 115 order statistics per column. One wave per column (8/WG),
// column cached in LDS; counting rank + rank->slot LDS map.
// ---------------------------------------------------------------------------
__global__ __launch_bounds__(256) void stage2_select_kernel(
    const float* __restrict__ corr2T, float* __restrict__ out) {
  __shared__ float colbuf[8][576];
  __shared__ int rmap[576];
  const int tid = threadIdx.x;
  const int w = tid >> 5, lane = tid & 31;
  const int col = blockIdx.x * 8 + w;
  const int b = col / 576, q = col % 576;

  for (int idx = tid; idx < 576; idx += 256) rmap[idx] = -1;
  __syncthreads();
  if (tid < 115) {
    const int rr = (int)rint(1.0 + (double)tid * (574.0 / 114.0));
    rmap[rr] = tid;
  }
  const float* src = corr2T + ((size_t)b * 576 + q) * 576;
  for (int idx = lane; idx < 576; idx += 32) colbuf[w][idx] = src[idx];
  __syncthreads();

  for (int i = lane; i < 576; i += 32) {
    const float vi = colbuf[w][i];
    int cnt = 0;
    for (int j = 0; j < 576; ++j) {
      const float vj = colbuf[w][j];
      cnt += (vj > vi) || (vj == vi && j < i);
    }
    const int rr = rmap[cnt];
    if (rr >= 0) out[((size_t)b * 115 + rr) * 576 + q] = vi;
  }
}

// ---------------------------------------------------------------------------
extern "C" void kernel_launch(void* const* d_in, const int* in_sizes, int n_in,
                              void* d_out, int out_size, void* d_ws,
                              size_t ws_size, hipStream_t stream) {
  const float* x = (const float*)d_in[0];
  float* out = (float*)d_out;
  float* ws = (float*)d_ws;

  float* pool = ws;                         // 8*64*576        = 294912 f
  float* inv2 = ws + 294912;                // 8*576           =   4608 f
  float* corr2T = ws + 294912 + 4608;       // 8*576*576       = 2654208 f

  stage1_kernel<<<512, 128, 0, stream>>>(x, pool);
  stage2_norm_kernel<<<18, 256, 0, stream>>>(pool, inv2);
  stage2_gram_kernel<<<72, 128, 0, stream>>>(pool, inv2, corr2T);
  stage2_select_kernel<<<576, 256, 0, stream>>>(corr2T, out);
}